// EdgeEncoding_3289944949216
// MI455X (gfx1250) — compile-verified
//
#include <hip/hip_runtime.h>
#include <hip/hip_bf16.h>

// ---------------------------------------------------------------------------
// EdgeEncoding on MI455X (gfx1250, wave32)
//
// Phase 1: S[e][l] = dot(edge_attr[e, :], edge_vector[l, :])  (E x 16 table)
//          computed as a (E x 32) x (32 x 16) GEMM via V_WMMA_F32_16X16X4_F32,
//          with rows of edge_attr that are exactly all -1.0 zeroed (the
//          reference's `valid` check). Table = E*16*4 B = 512 KB -> L2 resident.
// Phase 2: per pair p, out[p] = (1/len) * sum_{l: path[p,l]>=0} S[path[p,l]][l]
//          (pure streaming read of 16 MB of indices + L2-resident gathers).
// ---------------------------------------------------------------------------

typedef __attribute__((ext_vector_type(2))) float v2f;
typedef __attribute__((ext_vector_type(8))) float v8f;

#define D_DIM 32
#define L_DIM 16

// One wave (32 lanes) computes one 16(edges) x 16(hops) f32 tile of S.
__global__ __launch_bounds__(32) void build_table_wmma(
    const float* __restrict__ edge_attr,    // (E, 32) row-major
    const float* __restrict__ edge_vector,  // (16, 32) row-major
    float* __restrict__ S)                  // (E, 16) row-major
{
    const int lane  = threadIdx.x;     // 0..31
    const int half  = lane >> 4;       // 0: K-offsets {0,1}; 1: K-offsets {2,3}
    const int m     = lane & 15;
    const int ebase = blockIdx.x * 16;

    // A-fragment source: row (ebase+m) of edge_attr.
    const float* arow = edge_attr + (size_t)(ebase + m) * D_DIM;
    // B-fragment source: B[K][N] = edge_vector[N][Kglobal] -> row m of edge_vector.
    const float* brow = edge_vector + (size_t)m * D_DIM;

    v8f c = {};
    bool part_allneg = true;   // this lane's half of the row is all -1.0?

#pragma unroll
    for (int kk = 0; kk < 8; ++kk) {          // K = 32 in chunks of 4
        const int k0 = kk * 4 + half * 2;
        v2f a, b;
        a[0] = arow[k0 + 0];
        a[1] = arow[k0 + 1];
        b[0] = brow[k0 + 0];
        b[1] = brow[k0 + 1];
        part_allneg = part_allneg && (a[0] == -1.0f) && (a[1] == -1.0f);
        // 8 args: (neg_a, A, neg_b, B, c_mod, C, reuse_a, reuse_b)
        c = __builtin_amdgcn_wmma_f32_16x16x4_f32(
                false, a, false, b, (short)0, c, false, false);
    }

    // Row e (= ebase+l) is all -1.0 iff both lane l and lane l+16 saw all -1.0.
    const unsigned int bal = __builtin_amdgcn_ballot_w32(part_allneg);

    // D layout: VGPR r -> rows M = r (lanes 0-15) and M = r+8 (lanes 16-31),
    // N = lane & 15.
#pragma unroll
    for (int r = 0; r < 8; ++r) {
        const int M = r + half * 8;
        const bool allneg = ((bal >> M) & 1u) && ((bal >> (M + 16)) & 1u);
        const float v = allneg ? 0.0f : c[r];
        S[(size_t)(ebase + M) * L_DIM + m] = v;
    }
}

// One thread per pair: read 16 indices (64 B coalesced), 16 branchless gathers
// from the L2-resident table, sum, divide by path length.
__global__ __launch_bounds__(256) void gather_paths(
    const int*  __restrict__ paths,   // (P, 16)
    const float* __restrict__ S,      // (E, 16)
    float* __restrict__ out,          // (P,)
    int P)
{
    const int p = blockIdx.x * blockDim.x + threadIdx.x;
    if (p >= P) return;

    const int4* pp = (const int4*)(paths + (size_t)p * L_DIM);
    int4 q0 = pp[0], q1 = pp[1], q2 = pp[2], q3 = pp[3];
    int idx[L_DIM] = { q0.x, q0.y, q0.z, q0.w,
                       q1.x, q1.y, q1.z, q1.w,
                       q2.x, q2.y, q2.z, q2.w,
                       q3.x, q3.y, q3.z, q3.w };

    float sum = 0.0f;
    int   cnt = 0;
#pragma unroll
    for (int l = 0; l < L_DIM; ++l) {
        const int  e     = idx[l];
        const bool valid = (e >= 0);
        const int  esafe = valid ? e : 0;
        // Unconditional load (idx 0 is harmless), select to zero -> all 16
        // gathers issue without exec-mask churn, maximizing MLP.
        const float v = S[(size_t)esafe * L_DIM + l];
        sum += valid ? v : 0.0f;
        cnt += valid ? 1 : 0;
    }
    out[p] = (cnt != 0) ? (sum / (float)cnt) : 0.0f;
}

extern "C" void kernel_launch(void* const* d_in, const int* in_sizes, int n_in,
                              void* d_out, int out_size, void* d_ws, size_t ws_size,
                              hipStream_t stream) {
    // Inputs (setup_inputs order): x (unused), edge_attr, edge_paths, edge_vector
    const float* edge_attr   = (const float*)d_in[1];
    const int*   edge_paths  = (const int*)d_in[2];
    const float* edge_vector = (const float*)d_in[3];
    float*       out         = (float*)d_out;

    const int E = in_sizes[1] / D_DIM;     // 8192
    const int P = in_sizes[2] / L_DIM;     // 512*512

    float* S = (float*)d_ws;               // E*16*4 = 512 KB scratch

    // Phase 1: E/16 waves, one 16x16 tile each.
    build_table_wmma<<<E / 16, 32, 0, stream>>>(edge_attr, edge_vector, S);

    // Phase 2: one thread per pair.
    const int threads = 256;
    const int blocks  = (P + threads - 1) / threads;
    gather_paths<<<blocks, threads, 0, stream>>>(edge_paths, S, out, P);
}